// TransitionDown_89283780149664
// MI455X (gfx1250) — compile-verified
//
#include <hip/hip_runtime.h>

#define B_   8
#define N_   8192
#define S_   2048
#define K_   16
#define DIN  64
#define DOUT 128

typedef float v2f __attribute__((ext_vector_type(2)));
typedef float v8f __attribute__((ext_vector_type(8)));

// ---------------------------------------------------------------------------
// 1) Farthest point sampling: one block per batch, coords + distance in regs.
// ---------------------------------------------------------------------------
#define FPS_T 512
#define FPS_P (N_ / FPS_T)   // 16 points per thread
#define FPS_W (FPS_T / 32)   // 16 waves

__global__ __launch_bounds__(FPS_T) void fps_kernel(const float* __restrict__ coords,
                                                    float* __restrict__ new_coords) {
  __shared__ float s_val[FPS_W];
  __shared__ int   s_idx[FPS_W];
  __shared__ float s_cent[3];
  __shared__ int   s_far;

  const int b    = blockIdx.x;
  const int tid  = threadIdx.x;
  const int lane = tid & 31;
  const int wave = tid >> 5;
  const float* cb = coords + (size_t)b * N_ * 3;

  float cx[FPS_P], cy[FPS_P], cz[FPS_P], dist[FPS_P];
#pragma unroll
  for (int j = 0; j < FPS_P; ++j) {
    int p = j * FPS_T + tid;
    cx[j] = cb[p * 3 + 0];
    cy[j] = cb[p * 3 + 1];
    cz[j] = cb[p * 3 + 2];
    dist[j] = 1e10f;
  }

  int far = 0;  // reference scan starts at point 0
  for (int s = 0; s < S_; ++s) {
    // owner thread broadcasts the centroid coords (unrolled -> no dyn reg idx)
#pragma unroll
    for (int j = 0; j < FPS_P; ++j) {
      if (j * FPS_T + tid == far) {
        s_cent[0] = cx[j]; s_cent[1] = cy[j]; s_cent[2] = cz[j];
      }
    }
    __syncthreads();
    const float fx = s_cent[0], fy = s_cent[1], fz = s_cent[2];
    if (tid == 0) {
      float* o = new_coords + ((size_t)b * S_ + s) * 3;
      o[0] = fx; o[1] = fy; o[2] = fz;
    }

    float bv = -1.0f; int bi = 0x7fffffff;
#pragma unroll
    for (int j = 0; j < FPS_P; ++j) {
      float dx = cx[j] - fx, dy = cy[j] - fy, dz = cz[j] - fz;
      float d  = dx * dx + dy * dy + dz * dz;
      float nd = fminf(dist[j], d);
      dist[j]  = nd;
      int p = j * FPS_T + tid;
      if (nd > bv || (nd == bv && p < bi)) { bv = nd; bi = p; }
    }
    // wave32 argmax reduction (lowest index on ties, matching argmax)
#pragma unroll
    for (int off = 16; off > 0; off >>= 1) {
      float ov = __shfl_down(bv, off, 32);
      int   oi = __shfl_down(bi, off, 32);
      if (ov > bv || (ov == bv && oi < bi)) { bv = ov; bi = oi; }
    }
    if (lane == 0) { s_val[wave] = bv; s_idx[wave] = bi; }
    __syncthreads();
    if (wave == 0) {
      float v = (lane < FPS_W) ? s_val[lane] : -1.0f;
      int   i = (lane < FPS_W) ? s_idx[lane] : 0x7fffffff;
#pragma unroll
      for (int off = 16; off > 0; off >>= 1) {
        float ov = __shfl_down(v, off, 32);
        int   oi = __shfl_down(i, off, 32);
        if (ov > v || (ov == v && oi < i)) { v = ov; i = oi; }
      }
      if (lane == 0) s_far = i;
    }
    __syncthreads();
    far = s_far;
  }
}

// ---------------------------------------------------------------------------
// 2) KNN: one thread per query, register-resident sorted top-16, coords tiled
//    through LDS.
// ---------------------------------------------------------------------------
#define KNN_T    128
#define KNN_TILE 2048

__global__ __launch_bounds__(KNN_T) void knn_kernel(const float* __restrict__ coords,
                                                    const float* __restrict__ new_coords,
                                                    int* __restrict__ knn_idx) {
  __shared__ float sx[KNN_TILE], sy[KNN_TILE], sz[KNN_TILE];
  const int b    = blockIdx.x / (S_ / KNN_T);
  const int qblk = blockIdx.x % (S_ / KNN_T);
  const int q    = qblk * KNN_T + threadIdx.x;

  const float* qc = new_coords + ((size_t)b * S_ + q) * 3;
  const float qx = qc[0], qy = qc[1], qz = qc[2];
  const float* cb = coords + (size_t)b * N_ * 3;

  float nd[K_]; int ni[K_];
#pragma unroll
  for (int t = 0; t < K_; ++t) { nd[t] = 1e30f; ni[t] = 0; }

  for (int tile = 0; tile < N_; tile += KNN_TILE) {
    __syncthreads();
    for (int t = threadIdx.x; t < KNN_TILE * 3; t += KNN_T) {
      int p = t / 3, c = t - p * 3;
      float v = cb[(size_t)(tile + p) * 3 + c];
      if (c == 0) sx[p] = v; else if (c == 1) sy[p] = v; else sz[p] = v;
    }
    __syncthreads();
    for (int p = 0; p < KNN_TILE; ++p) {
      float dx = sx[p] - qx, dy = sy[p] - qy, dz = sz[p] - qz;
      float d  = dx * dx + dy * dy + dz * dz;
      if (d < nd[K_ - 1]) {                 // rarely taken after warm-up
        float vd = d; int vi = tile + p;
#pragma unroll
        for (int t = 0; t < K_; ++t) {      // bubble-through insertion
          if (vd < nd[t]) {
            float td = nd[t]; int ti = ni[t];
            nd[t] = vd; ni[t] = vi;
            vd = td; vi = ti;
          }
        }
      }
    }
  }
  int* o = knn_idx + ((size_t)b * S_ + q) * K_;
#pragma unroll
  for (int t = 0; t < K_; ++t) o[t] = ni[t];
}

// ---------------------------------------------------------------------------
// 3) Pointwise MLP via V_WMMA_F32_16X16X4_F32: h = relu((F·Wᵀ + b) BN-folded)
//    Each wave computes a 16(M)×128(N) tile; W staged in LDS (32 KB).
// ---------------------------------------------------------------------------
#define MLP_T 256

__global__ __launch_bounds__(MLP_T) void mlp_kernel(const float* __restrict__ feat,
                                                    const float* __restrict__ W,
                                                    const float* __restrict__ bias,
                                                    const float* __restrict__ gamma,
                                                    const float* __restrict__ beta,
                                                    const float* __restrict__ rmean,
                                                    const float* __restrict__ rvar,
                                                    float* __restrict__ h) {
  __shared__ float sW[DOUT * DIN];
  __shared__ float sSc[DOUT], sSh[DOUT];
  const int tid = threadIdx.x;

  for (int i = tid; i < DOUT * DIN; i += MLP_T) sW[i] = W[i];
  if (tid < DOUT) {
    float sc = gamma[tid] * rsqrtf(rvar[tid] + 1e-5f);
    sSc[tid] = sc;
    sSh[tid] = beta[tid] + (bias[tid] - rmean[tid]) * sc;
  }
  __syncthreads();

  const int wave = tid >> 5, lane = tid & 31;
  const int lo = lane & 15, hi = lane >> 4;        // A: M=lo, K-half = hi
  const int rowbase = blockIdx.x * 128 + wave * 16;
  const float* A = feat + (size_t)(rowbase + lo) * DIN;

  v8f acc[8];
#pragma unroll
  for (int g = 0; g < 8; ++g) acc[g] = (v8f){0, 0, 0, 0, 0, 0, 0, 0};

  for (int kk = 0; kk < DIN; kk += 4) {
    v2f a;                                          // 16x4 f32 A fragment
    a.x = A[kk + hi * 2 + 0];
    a.y = A[kk + hi * 2 + 1];
#pragma unroll
    for (int g = 0; g < 8; ++g) {
      int n = g * 16 + lo;                          // 4x16 f32 B fragment (Wᵀ)
      v2f bb;
      bb.x = sW[n * DIN + kk + hi * 2 + 0];
      bb.y = sW[n * DIN + kk + hi * 2 + 1];
      acc[g] = __builtin_amdgcn_wmma_f32_16x16x4_f32(
          /*neg_a=*/false, a, /*neg_b=*/false, bb,
          /*c_mod=*/(short)0, acc[g], /*reuse_a=*/false, /*reuse_b=*/false);
    }
  }

#pragma unroll
  for (int g = 0; g < 8; ++g) {
    int col = g * 16 + lo;
    float sc = sSc[col], sh = sSh[col];
#pragma unroll
    for (int r = 0; r < 8; ++r) {                   // C VGPR r: M=r (+8 for hi)
      int row = rowbase + r + hi * 8;
      float y = acc[g][r] * sc + sh;
      y = fmaxf(y, 0.0f);
      h[(size_t)row * DOUT + col] = y;
    }
  }
}

// ---------------------------------------------------------------------------
// 4) Gather K neighbors + channel-wise max-pool (h is L2-resident).
// ---------------------------------------------------------------------------
__global__ __launch_bounds__(256) void group_kernel(const float* __restrict__ h,
                                                    const int* __restrict__ knn_idx,
                                                    float* __restrict__ new_feat) {
  const int tid = threadIdx.x;
  const int c = tid & 127;
  const int q = blockIdx.x * 2 + (tid >> 7);
  const int b = q / S_;
  const int* idx  = knn_idx + (size_t)q * K_;
  const float* hb = h + (size_t)b * N_ * DOUT;
  float m = -1e30f;
#pragma unroll
  for (int k = 0; k < K_; ++k) {
    int id = idx[k];
    m = fmaxf(m, hb[(size_t)id * DOUT + c]);
  }
  new_feat[(size_t)q * DOUT + c] = m;
}

// ---------------------------------------------------------------------------
extern "C" void kernel_launch(void* const* d_in, const int* in_sizes, int n_in,
                              void* d_out, int out_size, void* d_ws, size_t ws_size,
                              hipStream_t stream) {
  (void)in_sizes; (void)n_in; (void)out_size; (void)ws_size;

  const float* coords = (const float*)d_in[0];
  const float* feat   = (const float*)d_in[1];
  const float* W      = (const float*)d_in[2];
  const float* bias   = (const float*)d_in[3];
  const float* gamma  = (const float*)d_in[4];
  const float* beta   = (const float*)d_in[5];
  const float* rmean  = (const float*)d_in[6];
  const float* rvar   = (const float*)d_in[7];

  float* new_coords = (float*)d_out;                         // [B,S,3]
  float* new_feat   = (float*)d_out + (size_t)B_ * S_ * 3;   // [B,S,128]

  int*   knn_idx = (int*)d_ws;                               // [B,S,K]   1 MB
  float* h = (float*)((char*)d_ws + (size_t)B_ * S_ * K_ * sizeof(int)); // [B,N,128] 33.5 MB

  fps_kernel  <<<B_,               FPS_T, 0, stream>>>(coords, new_coords);
  knn_kernel  <<<B_ * (S_/KNN_T),  KNN_T, 0, stream>>>(coords, new_coords, knn_idx);
  mlp_kernel  <<<(B_ * N_) / 128,  MLP_T, 0, stream>>>(feat, W, bias, gamma, beta,
                                                       rmean, rvar, h);
  group_kernel<<<(B_ * S_) / 2,    256,   0, stream>>>(h, knn_idx, new_feat);
}